// LocationSensitiveAttention_34437047779863
// MI455X (gfx1250) — compile-verified
//
#include <hip/hip_runtime.h>
#include <math.h>

typedef float v2f __attribute__((ext_vector_type(2)));
typedef float v8f __attribute__((ext_vector_type(8)));

#define B_SZ   64
#define T_SZ   2000
#define E_SZ   512
#define D_SZ   1024
#define A_SZ   128
#define K_SZ   31
#define PAD    15
#define TCHUNK 128          // t-values per workgroup in scores kernel
#define NWAVE  8            // 256 threads / wave32

#define NCHUNK_CTX 20
#define CTX_TCHUNK 100      // 20 * 100 = 2000

// workspace layout (in floats)
#define WS_PQ      0
#define WS_SCORES  (WS_PQ + B_SZ * A_SZ)
#define WS_PARTIAL (WS_SCORES + B_SZ * T_SZ)

// ---------------------------------------------------------------------------
// Kernel 1: pq[b][a] = sum_d query[b][d] * Wq[a][d]      (64 x 128, K=1024)
// ---------------------------------------------------------------------------
__global__ __launch_bounds__(A_SZ)
void pq_kernel(const float* __restrict__ query,
               const float* __restrict__ Wq,
               float* __restrict__ pq) {
    __shared__ float q[D_SZ];
    const int b   = blockIdx.x;
    const int tid = threadIdx.x;           // 128 threads, one per a
    for (int i = tid; i < D_SZ; i += A_SZ) q[i] = query[(size_t)b * D_SZ + i];
    __syncthreads();
    const float* wr = Wq + (size_t)tid * D_SZ;
    float acc = 0.f;
    for (int d = 0; d < D_SZ; d += 4) {
        float4 w = *(const float4*)(wr + d);
        acc += w.x * q[d] + w.y * q[d + 1] + w.z * q[d + 2] + w.w * q[d + 3];
    }
    pq[(size_t)b * A_SZ + tid] = acc;
}

// ---------------------------------------------------------------------------
// Kernel 2: scores[b][t] = v . tanh(pq[b] + memory[b,t,:]@Wm^T + conv)
//   grid = (ceil(T/128), B), block = 256 (8 waves). Wave w handles a
//   16t x 128a tile via V_WMMA_F32_16X16X4_F32, conv fused in epilogue.
// ---------------------------------------------------------------------------
__global__ __launch_bounds__(256)
void scores_kernel(const float* __restrict__ memory,
                   const float* __restrict__ awc,     // [B,T,2]
                   const float* __restrict__ Wm,      // [A,E]
                   const float* __restrict__ Wloc,    // [A,2,K]
                   const float* __restrict__ vvec,    // [A]
                   const float* __restrict__ pq,      // [B,A]
                   float* __restrict__ scores) {      // [B,T]
    __shared__ float wl[A_SZ * 2 * K_SZ];             // 7936 floats
    __shared__ float xs[2][TCHUNK + 2 * PAD];         // 2 x 158

    const int b     = blockIdx.y;
    const int chunk = blockIdx.x;
    const int tid   = threadIdx.x;
    const int w     = tid >> 5;        // wave id 0..7
    const int l     = tid & 31;        // lane
    const int half  = l >> 4;          // 0 / 1
    const int lm    = l & 15;          // lane-in-half

    // ---- stage Wloc and zero-padded x window into LDS ----
    for (int i = tid; i < A_SZ * 2 * K_SZ; i += 256) wl[i] = Wloc[i];
    const int tbase = chunk * TCHUNK;
    if (tid < TCHUNK + 2 * PAD) {
        const int tg = tbase - PAD + tid;
        float x0 = 0.f, x1 = 0.f;
        if (tg >= 0 && tg < T_SZ) {
            const float* xp = awc + ((size_t)b * T_SZ + tg) * 2;
            x0 = xp[0]; x1 = xp[1];
        }
        xs[0][tid] = x0;
        xs[1][tid] = x1;
    }
    __syncthreads();

    // ---- WMMA GEMM: pm tile [16t x 128a], K = 512, f32 16x16x4 ----
    const int t0w  = tbase + w * 16;
    const int trow = t0w + lm;
    const int tcl  = trow < T_SZ ? trow : T_SZ - 1;   // clamp (values masked)
    const float* mrow  = memory + ((size_t)b * T_SZ + tcl) * E_SZ + 2 * half;
    const float* wmrow = Wm + (size_t)lm * E_SZ + 2 * half;

    v8f c[8] = {};
    for (int e0 = 0; e0 < E_SZ; e0 += 4) {
        v2f afrag = *(const v2f*)(mrow + e0);
        #pragma unroll
        for (int nt = 0; nt < 8; ++nt) {
            v2f bfrag = *(const v2f*)(wmrow + (size_t)nt * 16 * E_SZ + e0);
            c[nt] = __builtin_amdgcn_wmma_f32_16x16x4_f32(
                false, afrag, false, bfrag, (short)0, c[nt], false, false);
        }
    }

    // ---- fused location conv + tanh + v-dot epilogue ----
    // this lane's 8 t-values are t = t0w + 8*half + r, r = 0..7
    // x window needed: xs[c][w*16 + 8*half + (r+k)], r+k in [0,37]
    float xw0[38], xw1[38];
    {
        const int xbase = w * 16 + 8 * half;
        #pragma unroll
        for (int i = 0; i < 38; ++i) {
            xw0[i] = xs[0][xbase + i];
            xw1[i] = xs[1][xbase + i];
        }
    }

    float tsum[8];
    #pragma unroll
    for (int r = 0; r < 8; ++r) tsum[r] = 0.f;

    #pragma unroll
    for (int nt = 0; nt < 8; ++nt) {
        const int   aidx = nt * 16 + lm;
        const float pqv  = pq[(size_t)b * A_SZ + aidx];
        const float vv   = vvec[aidx];
        float wreg[2 * K_SZ];
        #pragma unroll
        for (int m = 0; m < 2 * K_SZ; ++m) wreg[m] = wl[aidx * 2 * K_SZ + m];
        #pragma unroll
        for (int r = 0; r < 8; ++r) {
            float conv = 0.f;
            #pragma unroll
            for (int k = 0; k < K_SZ; ++k)
                conv += xw0[r + k] * wreg[k] + xw1[r + k] * wreg[K_SZ + k];
            const float val = c[nt][r] + pqv + conv;
            tsum[r] += tanhf(val) * vv;
        }
    }

    // reduce over the 16 lanes of each half (a-dimension), then store
    #pragma unroll
    for (int r = 0; r < 8; ++r) {
        float s = tsum[r];
        s += __shfl_xor(s, 1);
        s += __shfl_xor(s, 2);
        s += __shfl_xor(s, 4);
        s += __shfl_xor(s, 8);
        const int tout = t0w + 8 * half + r;
        if (lm == 0 && tout < T_SZ)
            scores[(size_t)b * T_SZ + tout] = s;
    }
}

// ---------------------------------------------------------------------------
// Kernel 3: softmax over T per batch row; writes alignments to d_out
// ---------------------------------------------------------------------------
__global__ __launch_bounds__(256)
void softmax_kernel(const float* __restrict__ scores,
                    float* __restrict__ align) {
    __shared__ float red[256];
    const int b = blockIdx.x, tid = threadIdx.x;
    const float* sr = scores + (size_t)b * T_SZ;
    float* ar = align + (size_t)b * T_SZ;

    float m = -1e30f;
    for (int t = tid; t < T_SZ; t += 256) m = fmaxf(m, sr[t]);
    red[tid] = m; __syncthreads();
    for (int s = 128; s > 0; s >>= 1) {
        if (tid < s) red[tid] = fmaxf(red[tid], red[tid + s]);
        __syncthreads();
    }
    const float mx = red[0];
    __syncthreads();

    float sum = 0.f;
    for (int t = tid; t < T_SZ; t += 256) {
        const float e = __expf(sr[t] - mx);
        ar[t] = e;
        sum += e;
    }
    red[tid] = sum; __syncthreads();
    for (int s = 128; s > 0; s >>= 1) {
        if (tid < s) red[tid] += red[tid + s];
        __syncthreads();
    }
    const float inv = 1.f / red[0];
    for (int t = tid; t < T_SZ; t += 256) ar[t] *= inv;
}

// ---------------------------------------------------------------------------
// Kernel 4: partial context over t-chunks (deterministic two-stage reduce)
// ---------------------------------------------------------------------------
__global__ __launch_bounds__(256)
void ctx_partial_kernel(const float* __restrict__ memory,
                        const float* __restrict__ align,
                        float* __restrict__ partial) {
    __shared__ float al[CTX_TCHUNK];
    const int ct = blockIdx.x, b = blockIdx.y, tid = threadIdx.x;
    if (tid < CTX_TCHUNK)
        al[tid] = align[(size_t)b * T_SZ + (size_t)ct * CTX_TCHUNK + tid];
    __syncthreads();

    float a0 = 0.f, a1 = 0.f;
    const float* mb = memory + ((size_t)b * T_SZ + (size_t)ct * CTX_TCHUNK) * E_SZ;
    for (int i = 0; i < CTX_TCHUNK; ++i) {
        const float av  = al[i];
        const float* row = mb + (size_t)i * E_SZ;
        a0 += av * row[tid];
        a1 += av * row[tid + 256];
    }
    float* p = partial + ((size_t)b * NCHUNK_CTX + ct) * E_SZ;
    p[tid]       = a0;
    p[tid + 256] = a1;
}

// ---------------------------------------------------------------------------
// Kernel 5: reduce partials -> context
// ---------------------------------------------------------------------------
__global__ __launch_bounds__(256)
void ctx_reduce_kernel(const float* __restrict__ partial,
                       float* __restrict__ ctx) {
    const int idx = blockIdx.x * 256 + threadIdx.x;     // 0 .. 32767
    const int b = idx >> 9, e = idx & (E_SZ - 1);
    float s = 0.f;
    for (int cc = 0; cc < NCHUNK_CTX; ++cc)
        s += partial[((size_t)b * NCHUNK_CTX + cc) * E_SZ + e];
    ctx[(size_t)b * E_SZ + e] = s;
}

// ---------------------------------------------------------------------------
extern "C" void kernel_launch(void* const* d_in, const int* in_sizes, int n_in,
                              void* d_out, int out_size, void* d_ws, size_t ws_size,
                              hipStream_t stream) {
    (void)in_sizes; (void)n_in; (void)out_size; (void)ws_size;
    const float* query  = (const float*)d_in[0];
    const float* memory = (const float*)d_in[1];
    const float* awc    = (const float*)d_in[2];
    const float* Wq     = (const float*)d_in[3];
    const float* Wm     = (const float*)d_in[4];
    const float* Wloc   = (const float*)d_in[5];
    const float* v      = (const float*)d_in[6];

    float* out   = (float*)d_out;
    float* ctx   = out;                    // [64, 512]
    float* align = out + B_SZ * E_SZ;      // [64, 2000]

    float* ws      = (float*)d_ws;
    float* pq      = ws + WS_PQ;
    float* scores  = ws + WS_SCORES;
    float* partial = ws + WS_PARTIAL;

    pq_kernel<<<dim3(B_SZ), dim3(A_SZ), 0, stream>>>(query, Wq, pq);
    scores_kernel<<<dim3((T_SZ + TCHUNK - 1) / TCHUNK, B_SZ), dim3(256), 0, stream>>>(
        memory, awc, Wm, Wloc, v, pq, scores);
    softmax_kernel<<<dim3(B_SZ), dim3(256), 0, stream>>>(scores, align);
    ctx_partial_kernel<<<dim3(NCHUNK_CTX, B_SZ), dim3(256), 0, stream>>>(memory, align, partial);
    ctx_reduce_kernel<<<dim3((B_SZ * E_SZ) / 256), dim3(256), 0, stream>>>(partial, ctx);
}